// HEEGNetStress_18889266168139
// MI455X (gfx1250) — compile-verified
//
#include <hip/hip_runtime.h>
#include <hip/hip_bf16.h>
#include <math.h>

typedef float v2f __attribute__((ext_vector_type(2)));
typedef float v8f __attribute__((ext_vector_type(8)));
typedef unsigned int u32x4 __attribute__((ext_vector_type(4)));
typedef int i32x4 __attribute__((ext_vector_type(4)));
typedef int i32x8 __attribute__((ext_vector_type(8)));

__device__ __forceinline__ float elu1(float x) { return x > 0.f ? x : (expf(x) - 1.f); }

// ---------------- sizes ----------------
constexpr int Bn = 128;      // batch
constexpr int HH = 32;       // conv1 H
constexpr int WX = 1024;     // input width
constexpr int W1 = 1025;     // conv1 / dw output width
constexpr int C2 = 32;       // main channel count
constexpr int W3c = 256;     // after pool/4
constexpr int W4c = 257;     // after ec1
constexpr int W5c = 258;     // after b2c1 / b2c2
constexpr int WLc = 64;      // lorentz W
constexpr int WPc = 16;      // lorentz W after pool
constexpr int CL = 33;       // lorentz dim

// ---------------- workspace offsets (floats) ----------------
constexpr long OF_SUM1 = 0;          // sum1[16], sq1[16]
constexpr long OF_SC1  = 32;         // scale1[16]
constexpr long OF_BI1  = 48;         // bias1[16]
constexpr long OF_SUM2 = 64;         // sum2[32], sq2[32]
constexpr long OF_SC2  = 128;
constexpr long OF_BI2  = 160;
constexpr long OF_ST3  = 192;        // sum3[32]@192, sq3[32]@224
constexpr long OF_SC3  = 256;
constexpr long OF_BI3  = 288;
constexpr long OF_SDOM = 320;        // 4*33
constexpr long OF_CNT  = 452;        // 4
constexpr long OF_MU   = 456;        // 4*33
constexpr long OF_VARS = 588;        // 4
constexpr long OF_VARF = 592;        // 4
constexpr long OF_STATS_END = 640;
constexpr long OF_H2  = 640;
constexpr long OF_H3  = OF_H2  + (long)Bn * C2 * W1;
constexpr long OF_H4  = OF_H3  + (long)Bn * C2 * W3c;
constexpr long OF_H5  = OF_H4  + (long)Bn * C2 * W4c;
constexpr long OF_H6  = OF_H5  + (long)Bn * C2 * W5c;
constexpr long OF_H8  = OF_H6  + (long)Bn * C2 * W5c;
constexpr long OF_VB  = OF_H8  + (long)Bn * WLc * CL;
constexpr long OF_H9  = OF_VB  + (long)Bn * WLc * CL;
constexpr long OF_H10 = OF_H9  + (long)Bn * WLc * CL;

// ---------------- init: zero stats region ----------------
__global__ void k_init(float* wsf) {
    for (int i = threadIdx.x; i < OF_STATS_END; i += blockDim.x) wsf[i] = 0.f;
}

// ---------------- K1: conv1 statistics (sum / sumsq per channel) ----------------
__global__ __launch_bounds__(256) void k_conv1_stats(const float* __restrict__ x,
                                                     const float* __restrict__ w1,
                                                     float* __restrict__ wsf) {
    __shared__ float w1s[16 * 32];
    __shared__ float acc[32];
    int tid = threadIdx.x;
    for (int i = tid; i < 512; i += 256) w1s[i] = w1[i];
    if (tid < 32) acc[tid] = 0.f;
    __syncthreads();
    float ls[16], lq[16];
    for (int c = 0; c < 16; ++c) { ls[c] = 0.f; lq[c] = 0.f; }
    const long total = (long)Bn * HH * W1;
    for (long p = (long)blockIdx.x * 256 + tid; p < total; p += (long)gridDim.x * 256) {
        int  wpos = (int)(p % W1);
        long bh   = p / W1;                       // b*32 + h
        const float* xr = x + bh * WX;
        float win[32];
        #pragma unroll
        for (int k = 0; k < 32; ++k) {
            int xi = wpos - 16 + k;
            win[k] = (xi >= 0 && xi < WX) ? xr[xi] : 0.f;
        }
        #pragma unroll
        for (int c = 0; c < 16; ++c) {
            float s = 0.f;
            #pragma unroll
            for (int k = 0; k < 32; ++k) s += w1s[c * 32 + k] * win[k];
            ls[c] += s; lq[c] += s * s;
        }
    }
    for (int c = 0; c < 16; ++c) {
        atomicAdd(&acc[c], ls[c]);
        atomicAdd(&acc[16 + c], lq[c]);
    }
    __syncthreads();
    if (tid < 32) atomicAdd(&wsf[OF_SUM1 + tid], acc[tid]);
}

// ---------------- K2: finalize BN1 ----------------
__global__ void k_fin_bn1(float* wsf, const float* bw, const float* bb) {
    int c = threadIdx.x;
    if (c >= 16) return;
    float cnt  = (float)Bn * HH * W1;
    float mean = wsf[OF_SUM1 + c] / cnt;
    float var  = wsf[OF_SUM1 + 16 + c] / cnt - mean * mean;
    float sc   = bw[c] * rsqrtf(var + 1e-3f);
    wsf[OF_SC1 + c] = sc;
    wsf[OF_BI1 + c] = bb[c] - mean * sc;
}

// ---------------- K3: fused conv1 + BN1 + depthwise-H conv  (+BN2 stats) ----------------
__global__ __launch_bounds__(256) void k_dwfuse(const float* __restrict__ x,
                                                const float* __restrict__ w1,
                                                const float* __restrict__ dw,
                                                float* __restrict__ wsf) {
    __shared__ float xs[32][160];
    __shared__ float w1s[16 * 32];
    __shared__ float dws[32 * 32];
    __shared__ float cs[16 * 128];
    __shared__ float s1[16], b1[16];
    __shared__ float acc2[64];
    int tid  = threadIdx.x;
    int b    = blockIdx.x / 9;
    int tile = blockIdx.x % 9;
    int w0   = tile * 128;
    for (int i = tid; i < 512;  i += 256) w1s[i] = w1[i];
    for (int i = tid; i < 1024; i += 256) dws[i] = dw[i];
    if (tid < 16) { s1[tid] = wsf[OF_SC1 + tid]; b1[tid] = wsf[OF_BI1 + tid]; }
    if (tid < 64) acc2[tid] = 0.f;
    for (int i = tid; i < 32 * 160; i += 256) {
        int h = i / 160, j = i % 160;
        int xi = w0 - 16 + j;
        xs[h][j] = (xi >= 0 && xi < WX) ? x[((long)b * HH + h) * WX + xi] : 0.f;
    }
    __syncthreads();
    int wl   = tid & 127;
    int ocb  = (tid >> 7) * 16;
    float accv[16];
    #pragma unroll
    for (int j = 0; j < 16; ++j) accv[j] = 0.f;
    for (int h = 0; h < 32; ++h) {
        for (int i = tid * 8; i < tid * 8 + 8; ++i) {        // 2048 conv values
            int g = i >> 7, wl2 = i & 127;
            float s = 0.f;
            const float* wr = &w1s[g * 32];
            const float* xr = &xs[h][wl2];
            #pragma unroll
            for (int k = 0; k < 32; ++k) s += wr[k] * xr[k];
            cs[g * 128 + wl2] = s * s1[g] + b1[g];
        }
        __syncthreads();
        #pragma unroll
        for (int j = 0; j < 16; ++j) {
            int oc = ocb + j;
            accv[j] += dws[oc * 32 + h] * cs[(oc >> 1) * 128 + wl];
        }
        __syncthreads();
    }
    int wg = w0 + wl;
    if (wg < W1) {
        float* H2 = wsf + OF_H2;
        for (int j = 0; j < 16; ++j) {
            int oc = ocb + j;
            float v = accv[j];
            H2[((long)b * C2 + oc) * W1 + wg] = v;
            atomicAdd(&acc2[oc], v);
            atomicAdd(&acc2[32 + oc], v * v);
        }
    }
    __syncthreads();
    if (tid < 64) atomicAdd(&wsf[OF_SUM2 + tid], acc2[tid]);
}

// ---------------- K4: finalize BN2 ----------------
__global__ void k_fin_bn2(float* wsf, const float* bw, const float* bb) {
    int c = threadIdx.x;
    if (c >= 32) return;
    float cnt  = (float)Bn * W1;                   // (B,32,1,1025): mean over B,H,W
    float mean = wsf[OF_SUM2 + c] / cnt;
    float var  = wsf[OF_SUM2 + 32 + c] / cnt - mean * mean;
    float sc   = bw[c] * rsqrtf(var + 1e-3f);
    wsf[OF_SC2 + c] = sc;
    wsf[OF_BI2 + c] = bb[c] - mean * sc;
}

// ---------------- K5: BN2 + ELU + avgpool/4 ----------------
__global__ __launch_bounds__(256) void k_bn2_elu_pool(float* wsf) {
    long idx = (long)blockIdx.x * 256 + threadIdx.x;
    if (idx >= (long)Bn * C2 * W3c) return;
    int wo = (int)(idx % W3c);
    int c  = (int)((idx / W3c) % C2);
    long bc = idx / W3c;                            // b*32 + c
    const float* src = wsf + OF_H2 + bc * W1 + wo * 4;
    float sc = wsf[OF_SC2 + c], bi = wsf[OF_BI2 + c];
    float s = 0.f;
    #pragma unroll
    for (int k = 0; k < 4; ++k) s += elu1(src[k] * sc + bi);
    wsf[OF_H3 + idx] = s * 0.25f;
}

// ---------------- K6: depthwise conv width-16 pad-8 (ec1): 256 -> 257 ----------------
__global__ __launch_bounds__(256) void k_ec1(const float* __restrict__ wgt, float* wsf) {
    long idx = (long)blockIdx.x * 256 + threadIdx.x;
    if (idx >= (long)Bn * C2 * W4c) return;
    int w  = (int)(idx % W4c);
    int c  = (int)((idx / W4c) % C2);
    long bc = idx / W4c;
    const float* src = wsf + OF_H3 + bc * W3c;
    float s = 0.f;
    #pragma unroll
    for (int k = 0; k < 16; ++k) {
        int xi = w - 8 + k;
        if (xi >= 0 && xi < W3c) s += wgt[c * 16 + k] * src[xi];
    }
    wsf[OF_H4 + idx] = s;
}

// ---------------- K7: depthwise conv width-16 pad-8 (b2c1): 257 -> 258 ----------------
__global__ __launch_bounds__(256) void k_b2c1(const float* __restrict__ wgt, float* wsf) {
    long idx = (long)blockIdx.x * 256 + threadIdx.x;
    if (idx >= (long)Bn * C2 * W5c) return;
    int w  = (int)(idx % W5c);
    int c  = (int)((idx / W5c) % C2);
    long bc = idx / W5c;
    const float* src = wsf + OF_H4 + bc * W4c;
    float s = 0.f;
    #pragma unroll
    for (int k = 0; k < 16; ++k) {
        int xi = w - 8 + k;
        if (xi >= 0 && xi < W4c) s += wgt[c * 16 + k] * src[xi];
    }
    wsf[OF_H5 + idx] = s;
}

// ---------------- K8: b2c2 pointwise 32x32 GEMM via f32 WMMA, B staged by TDM ----------------
// One block per batch b.  Tensor Data Mover DMAs the 32x258 f32 activation slab into LDS
// (TENSORcnt-tracked, EXEC-independent), then 4 waves sweep the 17 N-tiles doing
// D(32x16) = A(32x32) x B(32x16) with V_WMMA_F32_16X16X4_F32 from LDS-resident operands.
// Fragment layouts per CDNA5 ISA 7.12.2 (f32, K=4): A VGPR0 = K{0,2}, VGPR1 = K{1,3};
// B symmetric (VGPR0 rows K0/K2 striped across lanes); C/D rows M=j / M=j+8.
__global__ __launch_bounds__(128) void k_b2c2_wmma(const float* __restrict__ w,
                                                   float* __restrict__ wsf) {
    __shared__ float Ws[32 * 32];
    __shared__ float Bs[C2 * W5c];                  // 32 x 258 slab (33 KB), TDM destination
    __shared__ float acc[64];                       // BN3 sum[32], sq[32]
    int tid = threadIdx.x;
    int b   = blockIdx.x;
    for (int i = tid; i < 1024; i += 128) Ws[i] = w[i];
    if (tid < 64) acc[tid] = 0.f;

    const float* slab = wsf + OF_H5 + (long)b * C2 * W5c;
    if (tid < 32) {                                 // wave 0 drives the TDM
        if (b + 1 < Bn) __builtin_prefetch(slab + (long)C2 * W5c, 0, 1);
        unsigned long long ga = (unsigned long long)slab;          // 57-bit global byte addr
        unsigned int ldsoff = (unsigned int)(unsigned long long)&Bs[0]; // LDS offset = addr[31:0]
        // D# group 0: count=1 | lds_addr | global_addr[56:0] | type=2 ("image")
        u32x4 g0;
        g0.x = 1u;
        g0.y = ldsoff;
        g0.z = (unsigned int)(ga & 0xFFFFFFFFu);
        g0.w = (unsigned int)((ga >> 32) & 0x1FFFFFFu) | (2u << 30);
        // D# group 1: data_size=4B; tensor_dim0=258, tensor_dim1=32;
        //             tile_dim0=258, tile_dim1=32; tensor_dim0_stride=258
        i32x8 g1;
        g1[0] = (2 << 16);                          // data_size = 2 -> 4 bytes
        g1[1] = (int)((unsigned)(W5c & 0xFFFF) << 16); // tensor_dim0[15:0] at bits[63:48]
        g1[2] = (int)((unsigned)(C2 & 0xFFFF) << 16);  // tensor_dim1[15:0] at bits[95:80]
        g1[3] = (int)((unsigned)(W5c & 0xFFFF) << 16); // tile_dim0 at bits[127:112]
        g1[4] = C2;                                 // tile_dim1 = 32, tile_dim2 = 0
        g1[5] = W5c;                                // tensor_dim0_stride low 32
        g1[6] = 0;
        g1[7] = 0;
        i32x4 gz = {};                              // groups 2/3 unused (2-D tile)
#if defined(__clang_major__) && (__clang_major__ >= 23)
        i32x8 gz8 = {};
        __builtin_amdgcn_tensor_load_to_lds(g0, g1, gz, gz, gz8, 0);
#else
        __builtin_amdgcn_tensor_load_to_lds(g0, g1, gz, gz, 0);
#endif
        __builtin_amdgcn_s_wait_tensorcnt(0);
    }
    __syncthreads();                                // slab + weights visible to all waves

    int wave = tid >> 5;
    int lane = tid & 31;
    int hi   = (lane >= 16) ? 1 : 0;
    int m    = lane & 15;
    float* H6 = wsf + OF_H6;

    for (int nt = wave; nt < 17; nt += 4) {         // 17 N-tiles of 16 columns
        int w0   = nt * 16;
        int wcol = w0 + m;
        bool colok = (wcol < W5c);
        v8f c0 = {}; v8f c1 = {};
        #pragma unroll
        for (int kk = 0; kk < 8; ++kk) {
            int k0 = kk * 4 + (hi ? 2 : 0);
            int k1 = k0 + 1;
            v2f a0, a1, bf;
            a0.x = Ws[m * 32 + k0];        a0.y = Ws[m * 32 + k1];
            a1.x = Ws[(m + 16) * 32 + k0]; a1.y = Ws[(m + 16) * 32 + k1];
            float bx = 0.f, by = 0.f;
            if (colok) { bx = Bs[k0 * W5c + wcol]; by = Bs[k1 * W5c + wcol]; }
            bf.x = bx; bf.y = by;
            c0 = __builtin_amdgcn_wmma_f32_16x16x4_f32(false, a0, false, bf, (short)0, c0, false, false);
            c1 = __builtin_amdgcn_wmma_f32_16x16x4_f32(false, a1, false, bf, (short)0, c1, false, false);
        }
        if (colok) {
            float* outb = H6 + (long)b * C2 * W5c + wcol;
            #pragma unroll
            for (int j = 0; j < 8; ++j) {
                int row = j + (hi ? 8 : 0);
                float v0 = c0[j], v1 = c1[j];
                outb[(long)row * W5c] = v0;
                outb[(long)(row + 16) * W5c] = v1;
                atomicAdd(&acc[row],           v0);
                atomicAdd(&acc[32 + row],      v0 * v0);
                atomicAdd(&acc[row + 16],      v1);
                atomicAdd(&acc[32 + row + 16], v1 * v1);
            }
        }
    }
    __syncthreads();
    if (tid < 64) atomicAdd(&wsf[OF_ST3 + tid], acc[tid]);
}

// ---------------- K9: finalize BN3 ----------------
__global__ void k_fin_bn3(float* wsf, const float* bw, const float* bb) {
    int c = threadIdx.x;
    if (c >= 32) return;
    float cnt  = (float)Bn * W5c;
    float mean = wsf[OF_ST3 + c] / cnt;
    float var  = wsf[OF_ST3 + 32 + c] / cnt - mean * mean;
    float sc   = bw[c] * rsqrtf(var + 1e-3f);
    wsf[OF_SC3 + c] = sc;
    wsf[OF_BI3 + c] = bb[c] - mean * sc;
}

// ---------------- K10: BN3+ELU+pool/4 + normalize + lift + 33x33 matmul + re-lift ----------------
__global__ __launch_bounds__(256) void k_lift_lc1(const float* __restrict__ lc1, float* wsf) {
    __shared__ float lcs[CL * CL];
    int tid = threadIdx.x;
    for (int i = tid; i < CL * CL; i += 256) lcs[i] = lc1[i];
    __syncthreads();
    int p = blockIdx.x * 256 + tid;                 // (b, wo), 128*64 total
    if (p >= Bn * WLc) return;
    int b = p >> 6, wo = p & 63;
    float hv[32];
    float n2 = 0.f;
    for (int c = 0; c < 32; ++c) {
        const float* src = wsf + OF_H6 + ((long)b * C2 + c) * W5c + wo * 4;
        float sc = wsf[OF_SC3 + c], bi = wsf[OF_BI3 + c];
        float s = 0.f;
        #pragma unroll
        for (int k = 0; k < 4; ++k) s += elu1(src[k] * sc + bi);
        s *= 0.25f;
        hv[c] = s;
        n2 += s * s;
    }
    float inv = 1.f / fmaxf(sqrtf(n2), 1e-12f);
    float hn2 = 0.f;
    for (int c = 0; c < 32; ++c) { hv[c] *= inv; hn2 += hv[c] * hv[c]; }
    float t = sqrtf(1.f + hn2);
    float ysum = 0.f;
    float ysv[32];
    for (int o = 1; o < CL; ++o) {
        float a = lcs[o * CL] * t;
        for (int c = 0; c < 32; ++c) a += lcs[o * CL + 1 + c] * hv[c];
        ysv[o - 1] = a;
        ysum += a * a;
    }
    float* dst = wsf + OF_H8 + (long)p * CL;
    dst[0] = sqrtf(1.f + ysum);
    for (int i = 0; i < 32; ++i) dst[1 + i] = ysv[i];
}

// ---------------- K11a: per-domain point sums ----------------
__global__ void k_dom_sums(const int* __restrict__ domains, float* wsf) {
    int b = blockIdx.x;
    int d = domains[b];
    int tid = threadIdx.x;
    if (tid < CL) {
        float s = 0.f;
        const float* src = wsf + OF_H8 + (long)b * WLc * CL + tid;
        for (int w = 0; w < WLc; ++w) s += src[(long)w * CL];
        atomicAdd(&wsf[OF_SDOM + d * CL + tid], s);
    } else if (tid == CL) {
        atomicAdd(&wsf[OF_CNT + d], (float)WLc);
    }
}

// ---------------- K11b: mu per domain ----------------
__global__ void k_dom_mu(float* wsf) {
    int d = threadIdx.x;
    if (d >= 4) return;
    float cn = fmaxf(wsf[OF_CNT + d], 1.f);
    float sb[CL];
    for (int c = 0; c < CL; ++c) sb[c] = wsf[OF_SDOM + d * CL + c] / cn;
    float m = -sb[0] * sb[0];
    for (int c = 1; c < CL; ++c) m += sb[c] * sb[c];
    float inv = rsqrtf(fmaxf(fabsf(m), 1e-8f));
    for (int c = 0; c < CL; ++c) wsf[OF_MU + d * CL + c] = sb[c] * inv;
}

// ---------------- K11c: log-map v, accumulate dist^2 ----------------
__global__ __launch_bounds__(256) void k_dbn_v(const int* __restrict__ domains, float* wsf) {
    int p = blockIdx.x * 256 + threadIdx.x;
    if (p >= Bn * WLc) return;
    int b = p / WLc;
    int d = domains[b];
    float pt[CL], mb[CL];
    const float* src = wsf + OF_H8 + (long)p * CL;
    for (int c = 0; c < CL; ++c) { pt[c] = src[c]; mb[c] = wsf[OF_MU + d * CL + c]; }
    float ip = -mb[0] * pt[0];
    for (int c = 1; c < CL; ++c) ip += mb[c] * pt[c];
    float dist = acoshf(fmaxf(-ip, 1.f + 1e-7f));
    float u[CL];
    for (int c = 0; c < CL; ++c) u[c] = pt[c] + ip * mb[c];
    float uu = -u[0] * u[0];
    for (int c = 1; c < CL; ++c) uu += u[c] * u[c];
    float un = sqrtf(fmaxf(uu, 1e-8f));
    float f = dist / un;
    float* dst = wsf + OF_VB + (long)p * CL;
    for (int c = 0; c < CL; ++c) dst[c] = f * u[c];
    atomicAdd(&wsf[OF_VARS + d], dist * dist);
}

// ---------------- K11d: variance factor per domain ----------------
__global__ void k_dom_varfac(float* wsf, const float* gamma) {
    int d = threadIdx.x;
    if (d >= 4) return;
    float var = wsf[OF_VARS + d] / fmaxf(wsf[OF_CNT + d], 1.f);
    wsf[OF_VARF + d] = gamma[0] * rsqrtf(var + 1e-5f);
}

// ---------------- K11e: transport, scale, exp-map, beta shift, lorentz ELU ----------------
__global__ __launch_bounds__(256) void k_dbn_y(const int* __restrict__ domains,
                                               const float* __restrict__ beta, float* wsf) {
    int p = blockIdx.x * 256 + threadIdx.x;
    if (p >= Bn * WLc) return;
    int b = p / WLc;
    int d = domains[b];
    float v[CL], mb[CL], be[CL];
    const float* src = wsf + OF_VB + (long)p * CL;
    for (int c = 0; c < CL; ++c) { v[c] = src[c]; mb[c] = wsf[OF_MU + d * CL + c]; be[c] = beta[c]; }
    float f1 = -v[0] / (1.f + mb[0]);
    for (int c = 0; c < CL; ++c) v[c] += f1 * (mb[c] + (c == 0 ? 1.f : 0.f));
    float fac = wsf[OF_VARF + d];
    for (int c = 0; c < CL; ++c) v[c] *= fac;
    float mbv = -be[0] * v[0];
    for (int c = 1; c < CL; ++c) mbv += be[c] * v[c];
    float f2 = mbv / (1.f + be[0]);
    for (int c = 0; c < CL; ++c) v[c] += f2 * (be[c] + (c == 0 ? 1.f : 0.f));
    float vv = -v[0] * v[0];
    for (int c = 1; c < CL; ++c) vv += v[c] * v[c];
    float vn = sqrtf(fmaxf(vv, 1e-8f));
    float ch = coshf(vn), sh = sinhf(vn) / vn;
    float* dst = wsf + OF_H9 + (long)p * CL;
    float ss = 0.f;
    for (int c = 1; c < CL; ++c) {
        float y = ch * be[c] + sh * v[c];
        float s = elu1(y);
        dst[c] = s;
        ss += s * s;
    }
    dst[0] = sqrtf(1.f + ss);
}

// ---------------- K12: lorentz avgpool/4 ----------------
__global__ __launch_bounds__(256) void k_lpool(float* wsf) {
    int q = blockIdx.x * 256 + threadIdx.x;
    if (q >= Bn * WPc) return;
    int b = q >> 4, wo = q & 15;
    float s[CL];
    const float* src = wsf + OF_H9 + ((long)b * WLc + wo * 4) * CL;
    for (int c = 0; c < CL; ++c)
        s[c] = 0.25f * (src[c] + src[CL + c] + src[2 * CL + c] + src[3 * CL + c]);
    float m = -s[0] * s[0];
    for (int c = 1; c < CL; ++c) m += s[c] * s[c];
    float inv = rsqrtf(fmaxf(fabsf(m), 1e-8f));
    float* dst = wsf + OF_H10 + (long)q * CL;
    for (int c = 0; c < CL; ++c) dst[c] = s[c] * inv;
}

// ---------------- K13: features + hyperbolic MLR head ----------------
__global__ __launch_bounds__(512) void k_head(const float* __restrict__ z,
                                              const float* __restrict__ mlr_a,
                                              float* __restrict__ out, float* wsf) {
    __shared__ float r1[512], r2[512], r3[512];
    int b = blockIdx.x, i = threadIdx.x;
    float fsv = wsf[OF_H10 + ((long)b * WPc + (i >> 5)) * CL + 1 + (i & 31)];
    float zv  = z[i];
    r1[i] = fsv * fsv; r2[i] = fsv * zv; r3[i] = zv * zv;
    out[128 + (long)b * 513 + 1 + i] = fsv;
    __syncthreads();
    for (int s = 256; s > 0; s >>= 1) {
        if (i < s) { r1[i] += r1[i + s]; r2[i] += r2[i + s]; r3[i] += r3[i + s]; }
        __syncthreads();
    }
    if (i == 0) {
        float ft = sqrtf(1.f + r1[0]);
        out[128 + (long)b * 513] = ft;
        float nz = sqrtf(r3[0]);
        float a  = mlr_a[0];
        float wt = sinhf(a) * nz;
        float cz = coshf(a) * nz;
        float bet = sqrtf(fmaxf(-wt * wt + cz * cz, 1e-8f));
        float alpha = -wt * ft + coshf(a) * r2[0];
        float dd = fabsf(asinhf(alpha / bet));
        float sg = (alpha > 0.f) ? 1.f : ((alpha < 0.f) ? -1.f : 0.f);
        out[b] = sg * bet * dd;
    }
}

// ---------------- launch ----------------
extern "C" void kernel_launch(void* const* d_in, const int* in_sizes, int n_in,
                              void* d_out, int out_size, void* d_ws, size_t ws_size,
                              hipStream_t stream) {
    const float* x       = (const float*)d_in[0];
    const int*   domains = (const int*)  d_in[1];
    const float* conv1_w = (const float*)d_in[2];
    const float* bn1_w   = (const float*)d_in[3];
    const float* bn1_b   = (const float*)d_in[4];
    const float* dw_w    = (const float*)d_in[5];
    const float* bn2_w   = (const float*)d_in[6];
    const float* bn2_b   = (const float*)d_in[7];
    const float* ec1_w   = (const float*)d_in[8];
    const float* b2c1_w  = (const float*)d_in[9];
    const float* b2c2_w  = (const float*)d_in[10];
    const float* bn3_w   = (const float*)d_in[11];
    const float* bn3_b   = (const float*)d_in[12];
    const float* lc1_w   = (const float*)d_in[13];
    const float* bn_beta = (const float*)d_in[14];
    const float* bn_gamma= (const float*)d_in[15];
    const float* mlr_a   = (const float*)d_in[16];
    const float* mlr_z   = (const float*)d_in[17];
    float* out = (float*)d_out;
    float* wsf = (float*)d_ws;

    k_init<<<1, 256, 0, stream>>>(wsf);
    k_conv1_stats<<<1024, 256, 0, stream>>>(x, conv1_w, wsf);
    k_fin_bn1<<<1, 32, 0, stream>>>(wsf, bn1_w, bn1_b);
    k_dwfuse<<<Bn * 9, 256, 0, stream>>>(x, conv1_w, dw_w, wsf);
    k_fin_bn2<<<1, 32, 0, stream>>>(wsf, bn2_w, bn2_b);
    {
        long n = (long)Bn * C2 * W3c;
        k_bn2_elu_pool<<<(int)((n + 255) / 256), 256, 0, stream>>>(wsf);
    }
    {
        long n = (long)Bn * C2 * W4c;
        k_ec1<<<(int)((n + 255) / 256), 256, 0, stream>>>(ec1_w, wsf);
    }
    {
        long n = (long)Bn * C2 * W5c;
        k_b2c1<<<(int)((n + 255) / 256), 256, 0, stream>>>(b2c1_w, wsf);
    }
    k_b2c2_wmma<<<Bn, 128, 0, stream>>>(b2c2_w, wsf);      // 1 block / batch, TDM-staged B
    k_fin_bn3<<<1, 32, 0, stream>>>(wsf, bn3_w, bn3_b);
    k_lift_lc1<<<(Bn * WLc) / 256, 256, 0, stream>>>(lc1_w, wsf);
    k_dom_sums<<<Bn, 64, 0, stream>>>(domains, wsf);
    k_dom_mu<<<1, 32, 0, stream>>>(wsf);
    k_dbn_v<<<(Bn * WLc) / 256, 256, 0, stream>>>(domains, wsf);
    k_dom_varfac<<<1, 32, 0, stream>>>(wsf, bn_gamma);
    k_dbn_y<<<(Bn * WLc) / 256, 256, 0, stream>>>(domains, bn_beta, wsf);
    k_lpool<<<(Bn * WPc + 255) / 256, 256, 0, stream>>>(wsf);
    k_head<<<Bn, 512, 0, stream>>>(mlr_z, mlr_a, out, wsf);
}